// srVAE_16243566313882
// MI455X (gfx1250) — compile-verified
//
#include <hip/hip_runtime.h>
#include <hip/hip_bf16.h>
#include <math.h>

// ---------------------------------------------------------------------------
// Barron robust-NLL integral, MI455X (gfx1250, wave32).
//  Kernel 1: per-feature parameter table (alpha/scale -> inv_scale, inv_beta,
//            k = alpha/2, c2 = (beta/alpha_safe)*log2(e)), 12288 x float4 in ws.
//  Kernel 2: per-pixel 100-sample integration of the two LIVE accumulators
//            (normal bin + high-edge bin; the low-edge one in the reference is
//            dead code). gt/pred staged to LDS with gfx1250 async-to-LDS DMA.
//  Hot math: v_exp_f32 / v_log_f32 via amdgcn builtins; bases & steps
//  pre-scaled by inv_scale so each sample residual is a single v_fma_f32.
// ---------------------------------------------------------------------------

#define NSAMP   100
#define DFEAT   12288

__device__ __constant__ constexpr float BIN_F  = 1.0f / 127.5f;
__device__ __constant__ constexpr float STEP_N = (1.0f / 127.5f) / (float)NSAMP;
__device__ __constant__ constexpr float STEP_E = (2.0f + 1.0f / 127.5f) / (float)NSAMP;

// -------------------------- per-feature parameters -------------------------
__global__ __launch_bounds__(256) void barron_params_kernel(
    const float* __restrict__ latent_alpha,
    const float* __restrict__ latent_scale,
    float4* __restrict__ P, int D)
{
    int d = blockIdx.x * blockDim.x + threadIdx.x;
    if (d >= D) return;

    // alpha = sigmoid(la)*(1.999-0.001)+0.001
    float la  = latent_alpha[d];
    float sig = 1.0f / (1.0f + expf(-la));
    float alpha = __builtin_fmaf(sig, 1.998f, 0.001f);

    // scale = (1 - 1e-5)*softplus(ls + ln(e-1))/1.0 + 1e-5   (stable softplus)
    float x  = latent_scale[d] + 0.54132485461292f;          // ln(e-1)
    float sp = fmaxf(x, 0.0f) + log1pf(expf(-fabsf(x)));
    float scale = __builtin_fmaf(1.0f - 1e-5f, sp, 1e-5f);

    float beta  = fmaxf(1e-6f, fabsf(alpha - 2.0f));
    float asafe = copysignf(fmaxf(1e-6f, fabsf(alpha)), alpha);
    float c     = beta / asafe;

    // exp(-loss) = exp2( c2 - c2 * (sq/beta + 1)^(alpha/2) ),  c2 = c*log2(e)
    P[d] = make_float4(1.0f / scale,                 // .x inv_scale
                       1.0f / beta,                  // .y inv_beta
                       0.5f * alpha,                 // .z k
                       c * 1.4426950408889634f);     // .w c2
}

// one sample of one stream, with pre-scaled base bs and step ss:
//   r = bs - fi*ss;  acc += exp2(c2 - c2 * (r^2*inv_beta + 1)^k)
// v_fma, v_fma(r*r*ib+1 via mul+fma), v_log_f32, v_mul, v_exp_f32, v_fma, v_exp_f32, v_add
#define ACC1(acc, bs, fi, ss, q)                                               \
    {                                                                          \
        float r_ = __builtin_fmaf(-(fi), (ss), (bs));                          \
        float t_ = __builtin_fmaf(r_ * r_, (q).y, 1.0f);                       \
        float u_ = __builtin_amdgcn_exp2f((q).z * __builtin_amdgcn_logf(t_));  \
        (acc) += __builtin_amdgcn_exp2f(__builtin_fmaf(-(q).w, u_, (q).w));    \
    }

// ------------------------------ main kernel --------------------------------
__global__ __launch_bounds__(256) void barron_nll_kernel(
    const float* __restrict__ gt,
    const float* __restrict__ pred,
    const float4* __restrict__ P,
    float* __restrict__ out, int N)
{
    __shared__ float4 smem[512];          // [0,256): gt tile, [256,512): pred tile

    const int tid  = threadIdx.x;
    int base       = blockIdx.x * 1024 + tid * 4;   // 4 floats per thread
    const bool act = (base < N);
    if (!act) base = 0;                             // clamp (keeps barrier uniform)

    // gfx1250 async DMA: global -> LDS (ASYNCcnt), 16B per lane per instruction.
    uint32_t goff = (uint32_t)base * 4u;                         // byte offset
    uint32_t ldsg = (uint32_t)(size_t)(&smem[0]) + (uint32_t)tid * 16u;
    uint32_t ldsp = ldsg + 4096u;
    asm volatile(
        "global_load_async_to_lds_b128 %0, %2, %4\n\t"
        "global_load_async_to_lds_b128 %1, %3, %5\n\t"
        "s_wait_asynccnt 0x0"
        :
        : "v"(ldsg), "v"(ldsp), "v"(goff), "v"(goff), "s"(gt), "s"(pred)
        : "memory");
    __syncthreads();

    float4 g = smem[tid];
    float4 p = smem[256 + tid];

    // feature index (base % 4 == 0, DFEAT % 4 == 0 -> 4 consecutive features)
    int d = base % DFEAT;
    float4 q0 = P[d + 0], q1 = P[d + 1], q2 = P[d + 2], q3 = P[d + 3];

    float d0 = p.x - g.x, d1 = p.y - g.y, d2 = p.z - g.z, d3 = p.w - g.w;

    // resid_n(i) = diff + BIN/2 - step/2   - i*step
    // resid_h(i) = diff + BIN   - step_e/2 - i*step_e
    // Pre-scale everything by inv_scale: one v_fma_f32 per sample residual.
    const float offn = 0.5f * BIN_F - 0.5f * STEP_N;
    const float offh = BIN_F - 0.5f * STEP_E;
    float bn0 = (d0 + offn) * q0.x, bn1 = (d1 + offn) * q1.x;
    float bn2 = (d2 + offn) * q2.x, bn3 = (d3 + offn) * q3.x;
    float bh0 = (d0 + offh) * q0.x, bh1 = (d1 + offh) * q1.x;
    float bh2 = (d2 + offh) * q2.x, bh3 = (d3 + offh) * q3.x;
    float sn0 = STEP_N * q0.x, sn1 = STEP_N * q1.x;
    float sn2 = STEP_N * q2.x, sn3 = STEP_N * q3.x;
    float sh0 = STEP_E * q0.x, sh1 = STEP_E * q1.x;
    float sh2 = STEP_E * q2.x, sh3 = STEP_E * q3.x;

    float an0 = 0.f, an1 = 0.f, an2 = 0.f, an3 = 0.f;   // normal-bin sums
    float ah0 = 0.f, ah1 = 0.f, ah2 = 0.f, ah3 = 0.f;   // high-edge sums

#pragma unroll 2
    for (int i = 0; i < NSAMP; ++i) {
        float fi = (float)i;
        // 8 independent transcendental chains -> deep trans-pipe overlap
        ACC1(an0, bn0, fi, sn0, q0); ACC1(ah0, bh0, fi, sh0, q0);
        ACC1(an1, bn1, fi, sn1, q1); ACC1(ah1, bh1, fi, sh1, q1);
        ACC1(an2, bn2, fi, sn2, q2); ACC1(ah2, bh2, fi, sh2, q2);
        ACC1(an3, bn3, fi, sn3, q3); ACC1(ah3, bh3, fi, sh3, q3);
    }

    if (act) {
        // step factors pulled out of the sums; only mask_upper is live.
        // ln(x) = v_log_f32(x) * ln(2)
        const float LN2 = 0.6931471805599453f;
        float4 r;
        r.x = LN2 * __builtin_amdgcn_logf((g.x == 1.0f) ? ah0 * STEP_E : an0 * STEP_N);
        r.y = LN2 * __builtin_amdgcn_logf((g.y == 1.0f) ? ah1 * STEP_E : an1 * STEP_N);
        r.z = LN2 * __builtin_amdgcn_logf((g.z == 1.0f) ? ah2 * STEP_E : an2 * STEP_N);
        r.w = LN2 * __builtin_amdgcn_logf((g.w == 1.0f) ? ah3 * STEP_E : an3 * STEP_N);
        *(float4*)(out + base) = r;
    }
}

// ------------------------------- launcher ----------------------------------
extern "C" void kernel_launch(void* const* d_in, const int* in_sizes, int n_in,
                              void* d_out, int out_size, void* d_ws, size_t ws_size,
                              hipStream_t stream)
{
    const float* gt   = (const float*)d_in[0];
    const float* pred = (const float*)d_in[1];
    const float* la   = (const float*)d_in[2];
    const float* ls   = (const float*)d_in[3];
    float*       out  = (float*)d_out;

    const int N = in_sizes[0];     // 512*3*64*64 = 6,291,456
    const int D = in_sizes[2];     // 12288

    float4* P = (float4*)d_ws;     // 12288 * 16B = 192 KB parameter table

    barron_params_kernel<<<(D + 255) / 256, 256, 0, stream>>>(la, ls, P, D);

    int blocks = (N + 1023) / 1024;   // 256 thr * 4 elems = 1024 elems/block
    barron_nll_kernel<<<blocks, 256, 0, stream>>>(gt, pred, P, out, N);
}